// EngramModule_46007689674957
// MI455X (gfx1250) — compile-verified
//
#include <hip/hip_runtime.h>
#include <math.h>

typedef float v2f __attribute__((ext_vector_type(2)));
typedef float v8f __attribute__((ext_vector_type(8)));

#define B_   16
#define T_   8192
#define G_   4
#define C_   32
#define D_   32
#define GC_  128
#define TT_  128                  // output tokens per block
#define HALO_ 16                  // >= (K-1)*dilation = 9, rounded to WMMA M
#define LT_  (TT_ + HALO_)        // 144 local tokens
#define NW_  9                    // 9 waves: one 16-token WMMA group each
#define NTHREADS_ (NW_ * 32)      // 288
#define KROW_ 164                 // keyb row stride (floats): 160 cols + bank pad
#define XROW_ 136                 // xbuf row stride (floats): 128 cols + bank pad
#define EPS_ 1.1920929e-07f

// LDS layout (float offsets)
#define OFF_SB    0                               // swizzled B fragments: 10*8*32 float2 = 5120
#define OFF_BIAS  (OFF_SB + 5120)                 // 160 (bk || bv, fragment order)
#define OFF_N1    (OFF_BIAS + 160)                // 128
#define OFF_N2    (OFF_N1 + 128)                  // 128
#define OFF_CN    (OFF_N2 + 128)                  // 128
#define OFF_CW    (OFF_CN + 128)                  // 512 (conv_w: 128 ch x 4 taps)
#define OFF_KEYB  (OFF_CW + 512)                  // 9*16*164 = 23616
#define OFF_XBUF  (OFF_KEYB + NW_ * 16 * KROW_)   // 144*136 = 19584
#define LDS_FLOATS (OFF_XBUF + LT_ * XROW_)       // 49376 floats = 197504 bytes

__global__ __launch_bounds__(NTHREADS_)
void engram_fused_kernel(const float* __restrict__ emb,
                         const float* __restrict__ hs,
                         const float* __restrict__ Wv,
                         const float* __restrict__ bv,
                         const float* __restrict__ Wk,
                         const float* __restrict__ bk,
                         const float* __restrict__ n1w,
                         const float* __restrict__ n2w,
                         const float* __restrict__ cnw,
                         const float* __restrict__ cw,
                         float* __restrict__ out)
{
    extern __shared__ float sm[];
    const int tid  = threadIdx.x;
    const int lane = tid & 31;
    const int wave = tid >> 5;
    const int row  = lane & 15;   // WMMA M/N index within half-wave
    const int hi   = lane >> 4;   // half-wave select

    const int b  = blockIdx.x / (T_ / TT_);
    const int t0 = (blockIdx.x % (T_ / TT_)) * TT_;

    // ---------------- stage weights into LDS ----------------
    // B fragments pre-swizzled: entry s=(j,kk,l) -> float2 = Wk/Wv[n=16*(j%8)+r][4kk+2h .. +1]
    for (int s = tid; s < 10 * 8 * 32; s += NTHREADS_) {
        int j = s >> 8, kk = (s >> 5) & 7, l = s & 31;
        int r = l & 15, h = l >> 4;
        int d = 4 * kk + 2 * h;
        float x0, x1;
        if (j < 8) { int n = 16 * j + r;      x0 = Wk[n * 32 + d]; x1 = Wk[n * 32 + d + 1]; }
        else       { int n = 16 * (j - 8) + r; x0 = Wv[n * 32 + d]; x1 = Wv[n * 32 + d + 1]; }
        sm[OFF_SB + 2 * s]     = x0;
        sm[OFF_SB + 2 * s + 1] = x1;
    }
    for (int s = tid; s < 160; s += NTHREADS_) {
        int j = s >> 4, r = s & 15;
        sm[OFF_BIAS + s] = (j < 8) ? bk[16 * j + r] : bv[16 * (j - 8) + r];
    }
    for (int s = tid; s < 128; s += NTHREADS_) {
        sm[OFF_N1 + s] = n1w[s];
        sm[OFF_N2 + s] = n2w[s];
        sm[OFF_CN + s] = cnw[s];
    }
    for (int s = tid; s < 512; s += NTHREADS_) sm[OFF_CW + s] = cw[s];

    // warm hidden_states for phase B while WMMA runs
    #pragma unroll
    for (int p = 0; p < 2; ++p) {
        int idx = tid + p * NTHREADS_;
        int tl = idx >> 2, g = idx & 3;
        int gt = t0 - HALO_ + tl;
        if (gt >= 0) __builtin_prefetch(hs + (((long)b * T_ + gt) * G_ + g) * C_, 0, 0);
    }
    __syncthreads();

    // ---------------- Phase A: fp32 WMMA projections ----------------
    // wave owns 16 local tokens [16*wave, 16*wave+16); D = A(16 tok x 32 emb) * B(32 x 160)
    {
        int gt = t0 - HALO_ + 16 * wave + row;
        long gtc = (gt < 0) ? 0 : gt;                     // halo clamp (rows zeroed later)
        const float* ep = emb + ((long)b * T_ + gtc) * D_ + 2 * hi;
        v2f a[8];
        #pragma unroll
        for (int kk = 0; kk < 8; ++kk)
            a[kk] = *(const v2f*)(ep + 4 * kk);           // A frag: M=row, K=4kk+2hi..+1

        const int kb = OFF_KEYB + wave * 16 * KROW_;
        #pragma unroll
        for (int j = 0; j < 10; ++j) {                    // j<8: key cols, j>=8: value cols
            float bias = sm[OFF_BIAS + 16 * j + row];
            v8f acc;
            #pragma unroll
            for (int v = 0; v < 8; ++v) acc[v] = bias;
            #pragma unroll
            for (int kk = 0; kk < 8; ++kk) {
                v2f bb = *(const v2f*)&sm[OFF_SB + ((j * 8 + kk) * 32 + lane) * 2];
                acc = __builtin_amdgcn_wmma_f32_16x16x4_f32(
                          false, a[kk], false, bb, (short)0, acc, false, false);
            }
            // C/D layout: vgpr v -> token v + 8*hi, col 16j + row
            #pragma unroll
            for (int v = 0; v < 8; ++v)
                sm[kb + (v + 8 * hi) * KROW_ + 16 * j + row] = acc[v];
        }
    }
    __syncthreads();

    // ---------------- Phase B: norms + gate + gated value + channel RMSNorm ----------------
    float gated[2][32];
    #pragma unroll
    for (int p = 0; p < 2; ++p) {
        int idx = tid + p * NTHREADS_;                    // 576 (token, g) pairs
        int tl = idx >> 2, g = idx & 3;
        int gt = t0 - HALO_ + tl;
        float* xb = &sm[OFF_XBUF + tl * XROW_ + g * C_];
        if (gt < 0) {                                     // zero-pad region of causal conv
            #pragma unroll
            for (int c = 0; c < 32; c += 4) *(float4*)&xb[c] = make_float4(0.f, 0.f, 0.f, 0.f);
            #pragma unroll
            for (int c = 0; c < 32; ++c) gated[p][c] = 0.f;
            continue;
        }
        const float* kp = &sm[OFF_KEYB + (tl >> 4) * 16 * KROW_ + (tl & 15) * KROW_];
        const float* hp = hs + (((long)b * T_ + gt) * G_ + g) * C_;
        float msk = 0.f, msq = 0.f, msv = 0.f, s2 = 0.f;
        #pragma unroll
        for (int c = 0; c < 32; c += 4) {
            float4 kv = *(const float4*)&kp[g * 32 + c];
            float4 hq = *(const float4*)&hp[c];
            float4 vv = *(const float4*)&kp[128 + c];
            float4 w1 = *(const float4*)&sm[OFF_N1 + g * 32 + c];
            float4 w2 = *(const float4*)&sm[OFF_N2 + g * 32 + c];
            msk += kv.x*kv.x + kv.y*kv.y + kv.z*kv.z + kv.w*kv.w;
            msq += hq.x*hq.x + hq.y*hq.y + hq.z*hq.z + hq.w*hq.w;
            msv += vv.x*vv.x + vv.y*vv.y + vv.z*vv.z + vv.w*vv.w;
            s2  += (kv.x*w1.x)*(hq.x*w2.x) + (kv.y*w1.y)*(hq.y*w2.y)
                 + (kv.z*w1.z)*(hq.z*w2.z) + (kv.w*w1.w)*(hq.w*w2.w);
        }
        float rsk = rsqrtf(msk * (1.f / 32.f) + EPS_);
        float rsq = rsqrtf(msq * (1.f / 32.f) + EPS_);
        float gp  = rsk * rsq * s2 * 0.17677669529663689f;       // /sqrt(32)
        float sg  = (gp > 0.f) ? 1.f : ((gp < 0.f) ? -1.f : 0.f);
        float gv  = sg * sqrtf(fmaxf(fabsf(gp), 1e-6f));
        float gate = 1.f / (1.f + __expf(-gv));
        float rsg  = rsqrtf(gate * gate * msv * (1.f / 32.f) + EPS_);
        #pragma unroll
        for (int c = 0; c < 32; c += 4) {
            float4 vv = *(const float4*)&kp[128 + c];
            float4 cnv = *(const float4*)&sm[OFF_CN + g * 32 + c];
            float4 gd, xn;
            gd.x = gate * vv.x; gd.y = gate * vv.y; gd.z = gate * vv.z; gd.w = gate * vv.w;
            gated[p][c]   = gd.x; gated[p][c+1] = gd.y;
            gated[p][c+2] = gd.z; gated[p][c+3] = gd.w;
            xn.x = gd.x * rsg * cnv.x; xn.y = gd.y * rsg * cnv.y;
            xn.z = gd.z * rsg * cnv.z; xn.w = gd.w * rsg * cnv.w;
            *(float4*)&xb[c] = xn;
        }
    }
    __syncthreads();

    // ---------------- Phase C: dilated causal depthwise conv + SiLU + residual ----------------
    #pragma unroll
    for (int p = 0; p < 2; ++p) {
        int idx = tid + p * NTHREADS_;
        int tl = idx >> 2, g = idx & 3;
        if (tl < HALO_) continue;                         // halo rows produce no output
        int gt = t0 - HALO_ + tl;
        float* op = out + (((long)b * T_ + gt) * G_ + g) * C_;
        const float* x0 = &sm[OFF_XBUF + (tl - 9) * XROW_ + g * C_];
        const float* x1 = &sm[OFF_XBUF + (tl - 6) * XROW_ + g * C_];
        const float* x2 = &sm[OFF_XBUF + (tl - 3) * XROW_ + g * C_];
        const float* x3 = &sm[OFF_XBUF + (tl    ) * XROW_ + g * C_];
        const float* w  = &sm[OFF_CW + g * C_ * 4];
        #pragma unroll
        for (int c = 0; c < 32; c += 4) {
            float4 r;
            #pragma unroll
            for (int cc = 0; cc < 4; ++cc) {
                int ch = c + cc;
                float y = w[ch * 4 + 0] * x0[ch] + w[ch * 4 + 1] * x1[ch]
                        + w[ch * 4 + 2] * x2[ch] + w[ch * 4 + 3] * x3[ch];
                y = y / (1.f + __expf(-y));               // SiLU
                float o = gated[p][ch] + y;
                if (cc == 0) r.x = o; else if (cc == 1) r.y = o;
                else if (cc == 2) r.z = o; else r.w = o;
            }
            *(float4*)&op[c] = r;
        }
    }
}

extern "C" void kernel_launch(void* const* d_in, const int* in_sizes, int n_in,
                              void* d_out, int out_size, void* d_ws, size_t ws_size,
                              hipStream_t stream) {
    const float* emb = (const float*)d_in[0];
    const float* hs  = (const float*)d_in[1];
    const float* Wv  = (const float*)d_in[2];
    const float* bv  = (const float*)d_in[3];
    const float* Wk  = (const float*)d_in[4];
    const float* bk  = (const float*)d_in[5];
    const float* n1w = (const float*)d_in[6];
    const float* n2w = (const float*)d_in[7];
    const float* cnw = (const float*)d_in[8];
    const float* cw  = (const float*)d_in[9];
    float* out = (float*)d_out;

    const size_t shmem = (size_t)LDS_FLOATS * sizeof(float);   // 197504 B < 320 KB/WGP
    (void)hipFuncSetAttribute((const void*)engram_fused_kernel,
                              hipFuncAttributeMaxDynamicSharedMemorySize, (int)shmem);

    dim3 grid(B_ * (T_ / TT_));     // 1024 blocks: (batch, 128-token tile)
    dim3 block(NTHREADS_);          // 9 waves of 32
    engram_fused_kernel<<<grid, block, shmem, stream>>>(
        emb, hs, Wv, bv, Wk, bk, n1w, n2w, cnw, cw, out);
    (void)in_sizes; (void)n_in; (void)out_size; (void)d_ws; (void)ws_size;
}